// scattering_2d_cov_21174188769979
// MI455X (gfx1250) — compile-verified
//
#include <hip/hip_runtime.h>
#include <math.h>

// Problem constants
#define NWV 32
#define NN  1024
#define MMC 1024
#define NM  (1024 * 1024)      // elements per plane
#define NSPLIT 2048            // K-splits for the gram pass
#define KCH (NM / NSPLIT)      // 512 K per split

typedef float v2f __attribute__((ext_vector_type(2)));
typedef float v8f __attribute__((ext_vector_type(8)));

// -------- 1024-point inverse FFT (radix-2 DIT) in LDS ------------------
// Input must be loaded at bit-reversed positions. No 1/N scaling here.
__device__ __forceinline__ void fft1024_inv(float2* a, int tid) {
  for (int s = 1; s <= 10; ++s) {
    const int half = 1 << (s - 1);
    #pragma unroll
    for (int t = 0; t < 2; ++t) {
      int b   = tid + 256 * t;          // butterfly id 0..511 (all disjoint)
      int j   = b & (half - 1);
      int blk = b >> (s - 1);
      int i0  = (blk << s) + j;
      int i1  = i0 + half;
      float ang = 6.28318530717958647692f * (float)j / (float)(1 << s); // +i: inverse
      float sw, cw;
      __sincosf(ang, &sw, &cw);
      float2 u = a[i0];
      float2 v = a[i1];
      float tr = v.x * cw - v.y * sw;
      float ti = v.x * sw + v.y * cw;
      a[i1] = make_float2(u.x - tr, u.y - ti);
      a[i0] = make_float2(u.x + tr, u.y + ti);
    }
    __syncthreads();
  }
}

// -------- Pass 1: complex multiply + inverse FFT along axis 2 (rows) ---
__global__ void k_cmul_ifft_rows(const float2* __restrict__ xhat,
                                 const float* __restrict__ pre,
                                 const float* __restrict__ pim,
                                 float2* __restrict__ mid) {
  __shared__ float2 a[1024];
  const int n = blockIdx.x, w = blockIdx.y, tid = threadIdx.x;
  const size_t rowx = (size_t)n * MMC;
  const size_t rowp = ((size_t)w * NN + n) * MMC;
  #pragma unroll
  for (int t = 0; t < 4; ++t) {
    int m = tid + 256 * t;
    float2 x = xhat[rowx + m];
    float p = pre[rowp + m];
    float q = pim[rowp + m];
    float cr = x.x * p - x.y * q;
    float ci = x.y * p + x.x * q;
    int r = (int)(__brev((unsigned)m) >> 22);   // 10-bit reverse
    a[r] = make_float2(cr, ci);
  }
  __syncthreads();
  fft1024_inv(a, tid);
  #pragma unroll
  for (int t = 0; t < 4; ++t) {
    int m = tid + 256 * t;
    mid[rowp + m] = a[m];
  }
}

// -------- Pass 2: inverse FFT along axis 1 (columns) + modulus ---------
// Column gathers (stride 8 KB) go straight into bit-reversed LDS slots via
// GLOBAL_LOAD_ASYNC_TO_LDS_B64 (ASYNCcnt), skipping the VGPR round-trip.
__global__ void k_ifft_cols_mod(const float2* __restrict__ mid,
                                float* __restrict__ modv) {
  __shared__ float2 a[1024];
  const int m = blockIdx.x, w = blockIdx.y, tid = threadIdx.x;
  const size_t base = (size_t)w * NN * MMC + m;
  #pragma unroll
  for (int t = 0; t < 4; ++t) {
    int n = tid + 256 * t;
    int r = (int)(__brev((unsigned)n) >> 22);
    // LDS byte offset = low 32 bits of the generic shared-pointer
    unsigned ldsa = (unsigned)(unsigned long long)(uintptr_t)&a[r];
    unsigned long long ga =
        (unsigned long long)(uintptr_t)(mid + base + (size_t)n * MMC);
    asm volatile("global_load_async_to_lds_b64 %0, %1, off"
                 :: "v"(ldsa), "v"(ga) : "memory");
  }
  asm volatile("s_wait_asynccnt 0" ::: "memory");
  __syncthreads();
  fft1024_inv(a, tid);
  const float sc = 1.0f / (1024.0f * 1024.0f);   // ifft2 normalization
  #pragma unroll
  for (int t = 0; t < 4; ++t) {
    int n = tid + 256 * t;
    float re = a[n].x * sc;
    float im = a[n].y * sc;
    modv[base + (size_t)n * MMC] = sqrtf(re * re + im * im + 1e-8f);
  }
}

// -------- Pass 3: all three Gram tiles via V_WMMA_F32_16X16X4_F32 ------
// One wave32 per block; rows 0..15 (A0) and 16..31 (A1) are loaded once per
// K-step and feed three WMMAs: (0,0)=A0*A0, (0,16)=A0*A1, (16,16)=A1*A1.
__device__ __forceinline__ v8f wmma_f32(v2f a, v2f b, v8f c) {
  return __builtin_amdgcn_wmma_f32_16x16x4_f32(
      /*neg_a=*/false, a, /*neg_b=*/false, b,
      /*c_mod=*/(short)0, c, /*reuse_a=*/false, /*reuse_b=*/false);
}

__global__ void k_gram_wmma(const float* __restrict__ modv,
                            float* __restrict__ partial) {
  const int split = blockIdx.x;
  const int lane  = threadIdx.x;
  const int mrow  = lane & 15;
  const int hi    = lane >> 4;
  const int koff  = hi << 1;                       // 0 or 2 (A 16x4 layout)
  const float* R0 = modv + (size_t)mrow * NM;        // rows 0..15
  const float* R1 = modv + (size_t)(16 + mrow) * NM; // rows 16..31
  const int k0 = split * KCH;

  v8f acc00 = {0.f,0.f,0.f,0.f,0.f,0.f,0.f,0.f};
  v8f acc01 = acc00, acc11 = acc00;
  for (int k = k0; k < k0 + KCH; k += 4) {
    if ((k & 31) == 0) {                            // prefetch next cachelines
      __builtin_prefetch(R0 + k + 64, 0, 0);
      __builtin_prefetch(R1 + k + 64, 0, 0);
    }
    v2f a0 = *(const v2f*)(R0 + k + koff);
    v2f a1 = *(const v2f*)(R1 + k + koff);
    acc00 = wmma_f32(a0, a0, acc00);   // tile (0,0)
    acc01 = wmma_f32(a0, a1, acc01);   // tile (0,16): A rows 0-15, B rows 16-31
    acc11 = wmma_f32(a1, a1, acc11);   // tile (16,16)
  }

  // C/D layout: VGPR r -> M = r (lanes 0-15) / r+8 (lanes 16-31), N = lane&15
  #pragma unroll
  for (int r = 0; r < 8; ++r) {
    const int e = (r + 8 * hi) * 16 + mrow;
    partial[((size_t)0 * NSPLIT + split) * 256 + e] = acc00[r];
    partial[((size_t)1 * NSPLIT + split) * 256 + e] = acc01[r];
    partial[((size_t)2 * NSPLIT + split) * 256 + e] = acc11[r];
  }
}

// -------- Pass 4: deterministic fixed-order reduction of K-splits ------
__global__ void k_gram_reduce(const float* __restrict__ partial,
                              float* __restrict__ gram) {
  const int tb = blockIdx.x;       // 0..2
  const int e  = threadIdx.x;      // 0..255 tile element
  const float* p = partial + (size_t)tb * NSPLIT * 256 + e;
  float s = 0.f;
  for (int i = 0; i < NSPLIT; ++i) s += p[(size_t)i * 256];
  const int mr = e >> 4, nc = e & 15;
  const int i0 = (tb == 2) ? 16 : 0;
  const int j0 = (tb == 0) ? 0 : 16;
  gram[(i0 + mr) * 32 + (j0 + nc)] = s * (1.0f / (1024.0f * 1024.0f));
}

// -------- Pass 5: extract triu indices (drop last (31,31)) -------------
__global__ void k_triu(const float* __restrict__ gram, float* __restrict__ out) {
  int t = blockIdx.x * blockDim.x + threadIdx.x;
  if (t >= 527) return;
  int i = 0, rem = t;
  while (rem >= 32 - i) { rem -= 32 - i; ++i; }
  int j = i + rem;
  out[t] = gram[i * 32 + j];
}

extern "C" void kernel_launch(void* const* d_in, const int* in_sizes, int n_in,
                              void* d_out, int out_size, void* d_ws, size_t ws_size,
                              hipStream_t stream) {
  (void)in_sizes; (void)n_in; (void)out_size; (void)ws_size;
  const float2* xhat = (const float2*)d_in[0];   // (1024,1024,2) f32
  const float*  pre  = (const float*)d_in[1];    // (32,1024,1024) f32
  const float*  pim  = (const float*)d_in[2];    // (32,1024,1024) f32
  float* out = (float*)d_out;                    // 527 f32

  char* ws = (char*)d_ws;
  float2* mid     = (float2*)ws;                                            // 256 MB
  float*  modv    = (float*)(ws + (size_t)256 * 1024 * 1024);               // 128 MB
  float*  partial = (float*)(ws + (size_t)384 * 1024 * 1024);               // 6 MB
  float*  gram    = (float*)(ws + (size_t)384 * 1024 * 1024 + 6291456);     // 4 KB

  dim3 g1(NN, NWV);
  k_cmul_ifft_rows<<<g1, 256, 0, stream>>>(xhat, pre, pim, mid);

  dim3 g2(MMC, NWV);
  k_ifft_cols_mod<<<g2, 256, 0, stream>>>(mid, modv);

  k_gram_wmma<<<NSPLIT, 32, 0, stream>>>(modv, partial);

  k_gram_reduce<<<3, 256, 0, stream>>>(partial, gram);

  k_triu<<<3, 256, 0, stream>>>(gram, out);
}